// attension_9216999817566
// MI455X (gfx1250) — compile-verified
//
#include <hip/hip_runtime.h>
#include <hip/hip_bf16.h>

#define N_ATOM 128
#define N_ATT  64

typedef __attribute__((ext_vector_type(2))) float v2f;
typedef __attribute__((ext_vector_type(8))) float v8f;

// ---------------------------------------------------------------------------
// Fused dual GEMM + bias:
//   att1 = feat @ W1 + b1 ; att2 = feat @ W2 + b2   (feat: n_nodes x 128)
// One wave32 per 16-row strip; the wave owns all 4 N-tiles of BOTH outputs
// (8 accumulator tiles = 64 VGPRs) so feat is read exactly once from HBM.
// K looped in steps of 4 with V_WMMA_F32_16X16X4_F32; each A fragment feeds
// 8 consecutive WMMAs.
//
// ISA fragment layout (wave32):
//   A (16x4): lanes 0-15 rows M=0..15, K={0,1} in v0/v1; lanes 16-31 K={2,3}.
//   B (4x16): lanes 0-15 cols N=0..15, K={0,1} in v0/v1; lanes 16-31 K={2,3}.
//   C/D (16x16): VGPR r -> row r (lanes 0-15) / row 8+r (lanes 16-31), col = lane%16.
// ---------------------------------------------------------------------------
__global__ __launch_bounds__(256) void gemm_bias_wmma_fused(
    const float* __restrict__ feat,
    const float* __restrict__ W1, const float* __restrict__ b1,
    const float* __restrict__ W2, const float* __restrict__ b2,
    float* __restrict__ att1, float* __restrict__ att2, int n_nodes)
{
  const int lane = threadIdx.x & 31;
  const int wave = blockIdx.x * (blockDim.x >> 5) + (threadIdx.x >> 5);
  const int m0   = wave * 16;
  if (m0 >= n_nodes) return;                      // wave-uniform guard
  const int half = lane >> 4;                     // 0: lanes 0-15, 1: lanes 16-31
  const int l    = lane & 15;

  int arow = m0 + l;
  if (arow >= n_nodes) arow = n_nodes - 1;        // clamp loads; stores guarded below
  const float* aptr = feat + (size_t)arow * N_ATOM + 2 * half;

  v8f acc1[4], acc2[4];
#pragma unroll
  for (int nt = 0; nt < 4; ++nt) {
    const float bv1 = b1[nt * 16 + l];
    const float bv2 = b2[nt * 16 + l];
#pragma unroll
    for (int i = 0; i < 8; ++i) { acc1[nt][i] = bv1; acc2[nt][i] = bv2; }
  }

  const int wOff = (2 * half) * N_ATT + l;        // + nt*16, + k*N_ATT

#pragma unroll 2
  for (int ks = 0; ks < N_ATOM / 4; ++ks) {
    // A fragment: feat[arow][4*ks + 2*half + {0,1}] — one b64 load, reused 8x
    v2f a = *(const v2f*)(aptr + 4 * ks);
    const float* w1p = W1 + (size_t)(4 * ks) * N_ATT + wOff;
    const float* w2p = W2 + (size_t)(4 * ks) * N_ATT + wOff;
#pragma unroll
    for (int nt = 0; nt < 4; ++nt) {
      v2f bb1, bb2;
      bb1.x = w1p[nt * 16];
      bb1.y = w1p[nt * 16 + N_ATT];
      bb2.x = w2p[nt * 16];
      bb2.y = w2p[nt * 16 + N_ATT];
      acc1[nt] = __builtin_amdgcn_wmma_f32_16x16x4_f32(false, a, false, bb1,
                                                       (short)0, acc1[nt], false, false);
      acc2[nt] = __builtin_amdgcn_wmma_f32_16x16x4_f32(false, a, false, bb2,
                                                       (short)0, acc2[nt], false, false);
    }
  }

  if (m0 + 16 <= n_nodes) {
    // Common path: full tile, no per-lane predication (keeps EXEC untouched).
#pragma unroll
    for (int r = 0; r < 8; ++r) {
      const size_t row = (size_t)(m0 + r + 8 * half);
#pragma unroll
      for (int nt = 0; nt < 4; ++nt) {
        att1[row * N_ATT + nt * 16 + l] = acc1[nt][r];
        att2[row * N_ATT + nt * 16 + l] = acc2[nt][r];
      }
    }
  } else {
#pragma unroll
    for (int r = 0; r < 8; ++r) {
      const int row = m0 + r + 8 * half;
      if (row < n_nodes) {
#pragma unroll
        for (int nt = 0; nt < 4; ++nt) {
          att1[(size_t)row * N_ATT + nt * 16 + l] = acc1[nt][r];
          att2[(size_t)row * N_ATT + nt * 16 + l] = acc2[nt][r];
        }
      }
    }
  }
}

// ---------------------------------------------------------------------------
__device__ __forceinline__ float swishf(float x) {
  return x * (1.0f / (1.0f + __expf(-x)));
}

__global__ __launch_bounds__(256) void init_seg(float* __restrict__ segmax,
                                                float* __restrict__ segsum,
                                                int n)
{
  int i = blockIdx.x * blockDim.x + threadIdx.x;
  if (i < n) {
    segmax[i] = -__builtin_inff();
    segsum[i] = 0.0f;
  }
}

// Pass 1: per-edge swish, atomic float-max into seg_max[dst]
__global__ __launch_bounds__(256) void edge_max_kernel(
    const float* __restrict__ att1, const float* __restrict__ att2,
    const int* __restrict__ src, const int* __restrict__ dst,
    float* __restrict__ segmax, int n_edges)
{
  long long gid = (long long)blockIdx.x * blockDim.x + threadIdx.x;
  int e = (int)(gid >> 4);
  if (e >= n_edges) return;
  int a = ((int)gid & 15) * 4;
  int sN = src[e], dN = dst[e];
  const float4 x1 = *(const float4*)(att1 + (size_t)sN * N_ATT + a);
  const float4 x2 = *(const float4*)(att2 + (size_t)dN * N_ATT + a);
  float* mp = segmax + (size_t)dN * N_ATT + a;
  float s0 = swishf(x1.x + x2.x);
  float s1 = swishf(x1.y + x2.y);
  float s2 = swishf(x1.z + x2.z);
  float s3 = swishf(x1.w + x2.w);
  __hip_atomic_fetch_max(mp + 0, s0, __ATOMIC_RELAXED, __HIP_MEMORY_SCOPE_AGENT);
  __hip_atomic_fetch_max(mp + 1, s1, __ATOMIC_RELAXED, __HIP_MEMORY_SCOPE_AGENT);
  __hip_atomic_fetch_max(mp + 2, s2, __ATOMIC_RELAXED, __HIP_MEMORY_SCOPE_AGENT);
  __hip_atomic_fetch_max(mp + 3, s3, __ATOMIC_RELAXED, __HIP_MEMORY_SCOPE_AGENT);
}

// Pass 2: e = exp(s - max); atomic float-add into seg_sum[dst]
__global__ __launch_bounds__(256) void edge_sum_kernel(
    const float* __restrict__ att1, const float* __restrict__ att2,
    const int* __restrict__ src, const int* __restrict__ dst,
    const float* __restrict__ segmax, float* __restrict__ segsum, int n_edges)
{
  long long gid = (long long)blockIdx.x * blockDim.x + threadIdx.x;
  int e = (int)(gid >> 4);
  if (e >= n_edges) return;
  int a = ((int)gid & 15) * 4;
  int sN = src[e], dN = dst[e];
  const float4 x1 = *(const float4*)(att1 + (size_t)sN * N_ATT + a);
  const float4 x2 = *(const float4*)(att2 + (size_t)dN * N_ATT + a);
  const float4 m  = *(const float4*)(segmax + (size_t)dN * N_ATT + a);
  float* sp = segsum + (size_t)dN * N_ATT + a;
  float e0 = __expf(swishf(x1.x + x2.x) - m.x);
  float e1 = __expf(swishf(x1.y + x2.y) - m.y);
  float e2 = __expf(swishf(x1.z + x2.z) - m.z);
  float e3 = __expf(swishf(x1.w + x2.w) - m.w);
  __hip_atomic_fetch_add(sp + 0, e0, __ATOMIC_RELAXED, __HIP_MEMORY_SCOPE_AGENT);
  __hip_atomic_fetch_add(sp + 1, e1, __ATOMIC_RELAXED, __HIP_MEMORY_SCOPE_AGENT);
  __hip_atomic_fetch_add(sp + 2, e2, __ATOMIC_RELAXED, __HIP_MEMORY_SCOPE_AGENT);
  __hip_atomic_fetch_add(sp + 3, e3, __ATOMIC_RELAXED, __HIP_MEMORY_SCOPE_AGENT);
}

// Pass 3: out = exp(s - max) / seg_sum[dst]   (recompute e; gathers are L2-resident)
__global__ __launch_bounds__(256) void edge_out_kernel(
    const float* __restrict__ att1, const float* __restrict__ att2,
    const int* __restrict__ src, const int* __restrict__ dst,
    const float* __restrict__ segmax, const float* __restrict__ segsum,
    float* __restrict__ out, int n_edges)
{
  long long gid = (long long)blockIdx.x * blockDim.x + threadIdx.x;
  int e = (int)(gid >> 4);
  if (e >= n_edges) return;
  int a = ((int)gid & 15) * 4;
  int sN = src[e], dN = dst[e];
  const float4 x1 = *(const float4*)(att1 + (size_t)sN * N_ATT + a);
  const float4 x2 = *(const float4*)(att2 + (size_t)dN * N_ATT + a);
  const float4 m  = *(const float4*)(segmax + (size_t)dN * N_ATT + a);
  const float4 ss = *(const float4*)(segsum + (size_t)dN * N_ATT + a);
  float4 o;
  o.x = __expf(swishf(x1.x + x2.x) - m.x) / ss.x;
  o.y = __expf(swishf(x1.y + x2.y) - m.y) / ss.y;
  o.z = __expf(swishf(x1.z + x2.z) - m.z) / ss.z;
  o.w = __expf(swishf(x1.w + x2.w) - m.w) / ss.w;
  *(float4*)(out + (size_t)e * N_ATT + a) = o;
}

// ---------------------------------------------------------------------------
extern "C" void kernel_launch(void* const* d_in, const int* in_sizes, int n_in,
                              void* d_out, int out_size, void* d_ws, size_t ws_size,
                              hipStream_t stream)
{
  const float* feat = (const float*)d_in[0];
  const int*   src  = (const int*)d_in[1];
  const int*   dst  = (const int*)d_in[2];
  const float* W1   = (const float*)d_in[3];
  const float* b1   = (const float*)d_in[4];
  const float* W2   = (const float*)d_in[5];
  const float* b2   = (const float*)d_in[6];

  const int n_nodes = in_sizes[0] / N_ATOM;
  const int n_edges = in_sizes[1];

  // Workspace: att1 | att2 | segmax | segsum  (4 x n_nodes*64 floats = 51.2 MB)
  float* att1   = (float*)d_ws;
  float* att2   = att1 + (size_t)n_nodes * N_ATT;
  float* segmax = att2 + (size_t)n_nodes * N_ATT;
  float* segsum = segmax + (size_t)n_nodes * N_ATT;

  // --- Fused WMMA GEMMs: one wave per 16-row strip (feat read exactly once)
  const int mTiles  = (n_nodes + 15) / 16;        // 3125 for n_nodes=50000
  const int gBlocks = (mTiles + 7) / 8;           // 8 waves per 256-thread block
  gemm_bias_wmma_fused<<<gBlocks, 256, 0, stream>>>(feat, W1, b1, W2, b2,
                                                    att1, att2, n_nodes);

  // --- init segment tables
  const int segN = n_nodes * N_ATT;
  init_seg<<<(segN + 255) / 256, 256, 0, stream>>>(segmax, segsum, segN);

  // --- edge passes: 16 threads (float4 each) per edge
  const long long nThreads = (long long)n_edges * 16;
  const int eBlocks = (int)((nThreads + 255) / 256);
  edge_max_kernel<<<eBlocks, 256, 0, stream>>>(att1, att2, src, dst, segmax, n_edges);
  edge_sum_kernel<<<eBlocks, 256, 0, stream>>>(att1, att2, src, dst, segmax, segsum, n_edges);
  edge_out_kernel<<<eBlocks, 256, 0, stream>>>(att1, att2, src, dst, segmax, segsum,
                                               (float*)d_out, n_edges);
}